// FourierEncoder_82815559401799
// MI455X (gfx1250) — compile-verified
//
#include <hip/hip_runtime.h>
#include <hip/hip_bf16.h>
#include <math.h>

// Problem constants (from reference)
#define Bn 4
#define Cn 8
#define Hn 128
#define Wn 256
#define En 32
#define NBn 4
#define FHn 65
#define FWn 129
#define HWn (Hn*Wn)            // 32768
#define FREQS (FHn*FWn)        // 8385

typedef float v2f __attribute__((ext_vector_type(2)));
typedef float v8f __attribute__((ext_vector_type(8)));

#define WMMA_F32(a, b, c) \
    __builtin_amdgcn_wmma_f32_16x16x4_f32(false, (a), false, (b), (short)0, (c), false, false)

// ---------------------------------------------------------------------------
// Generic batched GEMM on V_WMMA_F32_16X16X4_F32.
// C[z][m][n] = sum_k A[z][m][k] * B[z][k][n]  (+bias[m]) (gelu) (+res[z][m][n])
// One wave computes a 16x32 tile (two 16x16 accumulators sharing the A frag);
// 4 waves per block cover 128 columns.
// f32 WMMA fragment layouts (ISA 7.12.2):
//   A 16x4 : lane<16 -> (M=lane, K=k0+{0,1}) ; lane>=16 -> (M=lane-16, K=k0+{2,3})
//   B 4x16 : lane<16 -> (N=lane, K=k0+{0,1}) ; lane>=16 -> (N=lane-16, K=k0+{2,3})
//   C 16x16: vgpr v, lane<16 -> (M=v,   N=lane) ; lane>=16 -> (M=v+8, N=lane-16)
// All lda used by this file are even and all A base pointers 8B-aligned, so
// the A fragment is one 8-byte (v2f) load.
// ---------------------------------------------------------------------------
template<bool BIAS, bool RES, bool GELU>
__global__ __launch_bounds__(128)
void gemm16(const float* __restrict__ A, int lda, long sA,
            const float* __restrict__ B, int ldb, long sB,
            float* __restrict__ C, int ldc, long sC,
            const float* __restrict__ bias,
            const float* __restrict__ res, long sR,
            int M, int N, int K)
{
    const int z = blockIdx.z;
    A += (long)z * sA;
    B += (long)z * sB;
    C += (long)z * sC;
    if (RES) res += (long)z * sR;

    const int wave = threadIdx.x >> 5;
    const int lane = threadIdx.x & 31;
    const int half = lane >> 4;     // selects K pair within a 4-wide K step
    const int l    = lane & 15;

    const int m0 = blockIdx.y * 16;
    const int n0 = (blockIdx.x * 4 + wave) * 32;
    if (n0 >= N || m0 >= M) return;   // wave-uniform exit; EXEC stays full

    const int m = m0 + l;
    const int Km = K & ~3;

    v8f acc0 = {}, acc1 = {};

    if (m0 + 16 <= M && n0 + 32 <= N) {
        // ---------------- fast path: fully interior tile, branch-free loop
        const float* Ap  = A + (long)m * lda + half * 2;
        const float* Bp0 = B + (long)(half * 2) * ldb + n0 + l;
        const float* Bp1 = Bp0 + 16;
        const long bstep = 4L * ldb;
#pragma unroll 2
        for (int k0 = 0; k0 < Km; k0 += 4) {
            const v2f a = *(const v2f*)Ap;
            v2f b0, b1;
            b0.x = Bp0[0];
            b0.y = Bp0[ldb];
            b1.x = Bp1[0];
            b1.y = Bp1[ldb];
            __builtin_prefetch(Bp0 + 4 * bstep, 0, 1);
            Ap += 4;
            Bp0 += bstep;
            Bp1 += bstep;
            acc0 = WMMA_F32(a, b0, acc0);
            acc1 = WMMA_F32(a, b1, acc1);
        }
        if (Km < K) {   // K tail (K = 130 / 258 cases)
            const int ka = Km + half * 2;
            v2f a, b0, b1;
            a.x  = (ka     < K) ? Ap[0]    : 0.f;
            a.y  = (ka + 1 < K) ? Ap[1]    : 0.f;
            b0.x = (ka     < K) ? Bp0[0]   : 0.f;
            b0.y = (ka + 1 < K) ? Bp0[ldb] : 0.f;
            b1.x = (ka     < K) ? Bp1[0]   : 0.f;
            b1.y = (ka + 1 < K) ? Bp1[ldb] : 0.f;
            acc0 = WMMA_F32(a, b0, acc0);
            acc1 = WMMA_F32(a, b1, acc1);
        }
    } else {
        // ---------------- boundary tile: fully guarded
        const bool mok = (m < M);
        const int n_0 = n0 + l;
        const int n_1 = n0 + 16 + l;
        const bool nok0 = (n_0 < N);
        const bool nok1 = (n_1 < N);
        for (int k0 = 0; k0 < K; k0 += 4) {
            const int ka = k0 + half * 2;
            v2f a, b0, b1;
            a.x  = (mok  && ka     < K) ? A[(long)m * lda + ka    ]     : 0.f;
            a.y  = (mok  && ka + 1 < K) ? A[(long)m * lda + ka + 1]     : 0.f;
            b0.x = (nok0 && ka     < K) ? B[(long)ka * ldb + n_0]       : 0.f;
            b0.y = (nok0 && ka + 1 < K) ? B[(long)(ka + 1) * ldb + n_0] : 0.f;
            b1.x = (nok1 && ka     < K) ? B[(long)ka * ldb + n_1]       : 0.f;
            b1.y = (nok1 && ka + 1 < K) ? B[(long)(ka + 1) * ldb + n_1] : 0.f;
            acc0 = WMMA_F32(a, b0, acc0);
            acc1 = WMMA_F32(a, b1, acc1);
        }
    }

    // ---------------- epilogue
#pragma unroll
    for (int t = 0; t < 2; ++t) {
        const int n = n0 + t * 16 + l;
        if (n < N) {
            const v8f acc = t ? acc1 : acc0;
#pragma unroll
            for (int v = 0; v < 8; ++v) {
                const int mr = m0 + half * 8 + v;
                if (mr < M) {
                    float val = acc[v];
                    if (BIAS) val += bias[mr];
                    if (GELU) val = 0.5f * val * (1.f + erff(val * 0.70710678118f));
                    if (RES)  val += res[(long)mr * ldc + n];
                    C[(long)mr * ldc + n] = val;
                }
            }
        }
    }
}

// ---------------------------------------------------------------------------
// Per-frequency complex channel mix:  out[b,d,f] = sum_c xf[b,c,f] * w[d,c,f]
// XF2/XM layout: [b*E + e][part(0=Re,1=Im)*FREQS][kh*FW + kw], batch stride 2*FREQS.
// One block = one frequency; 128 threads = (b in 0..3) x (d in 0..31).
// Pure streaming of 68 MB of filter weights per FNO block -> HBM bound.
// ---------------------------------------------------------------------------
__global__ __launch_bounds__(128)
void mix_kernel(const float* __restrict__ XF2,
                const float* __restrict__ wr,
                const float* __restrict__ wi,
                float* __restrict__ XM)
{
    const int f = blockIdx.x;          // 0..FREQS-1
    const int t = threadIdx.x;         // 0..127
    const int d = t & 31;
    const int b = t >> 5;

    __shared__ float xre[Bn][En], xim[Bn][En];
    __shared__ float wre[En][En], wim[En][En];

    // stage xf: 4b x 32c complex (exactly 128 pairs)
    {
        const int bb = t >> 5, cc = t & 31;
        const long base = (long)(bb * En + cc) * (2 * FREQS) + f;
        xre[bb][cc] = XF2[base];
        xim[bb][cc] = XF2[base + FREQS];
    }
    // stage w: 32x32 complex, cooperative
    for (int idx = t; idx < En * En; idx += 128) {
        const int dd = idx >> 5, cc = idx & 31;
        const long base = (long)(dd * En + cc) * (long)FREQS + f;
        wre[dd][cc] = wr[base];
        wim[dd][cc] = wi[base];
    }
    __syncthreads();

    float outr = 0.f, outi = 0.f;
#pragma unroll 8
    for (int c = 0; c < En; ++c) {
        const float xr = xre[b][c], xi = xim[b][c];
        const float ar = wre[d][c], ai = wim[d][c];
        outr += xr * ar - xi * ai;
        outi += xr * ai + xi * ar;
    }
    const long ob = (long)(b * En + d) * (2 * FREQS) + f;
    XM[ob]         = outr;
    XM[ob + FREQS] = outi;
}

// ---------------------------------------------------------------------------
// DFT twiddle tables (norm='ortho' folded in: 1/16 per W pass, 1/sqrt(128) per H pass)
// ---------------------------------------------------------------------------
#define TWO_PI 6.28318530717958647692f
#define INV_SQRT_H 0.08838834764831844f   /* 1/sqrt(128) */
#define INV_SQRT_W 0.0625f                /* 1/sqrt(256) */

// rfft along W:  X[f] = (1/16) sum_w x[w] e^{-2pi i w f/256}
__global__ void build_wf(float* __restrict__ c, float* __restrict__ s)
{
    int i = blockIdx.x * 256 + threadIdx.x;
    if (i >= Wn * FWn) return;
    int w = i / FWn, f = i % FWn;
    float th = TWO_PI * (float)((w * f) & (Wn - 1)) / (float)Wn;
    c[i] =  cosf(th) * INV_SQRT_W;
    s[i] = -sinf(th) * INV_SQRT_W;
}

// fft along H (complex in, rows truncated to k<65), stacked-K tables [65][256]:
//   Yre = sum_h XRe*c + XIm*s ;  Yim = sum_h XIm*c - XRe*s ; th = 2pi h k/128
__global__ void build_hf(float* __restrict__ re, float* __restrict__ im)
{
    int i = blockIdx.x * 256 + threadIdx.x;
    if (i >= FHn * (2 * Hn)) return;
    int k = i / (2 * Hn), j = i % (2 * Hn);
    int h = j & (Hn - 1);
    float th = TWO_PI * (float)((h * k) & (Hn - 1)) / (float)Hn;
    float cv = cosf(th) * INV_SQRT_H, sv = sinf(th) * INV_SQRT_H;
    re[i] = (j < Hn) ?  cv : sv;
    im[i] = (j < Hn) ? -sv : cv;
}

// ifft along H (zero-padded k>=65), stacked-K tables [128][130]:
//   yre = sum_k YRe*c - YIm*s ; yim = sum_k YRe*s + YIm*c ; th = 2pi h k/128
__global__ void build_hi(float* __restrict__ re, float* __restrict__ im)
{
    int i = blockIdx.x * 256 + threadIdx.x;
    if (i >= Hn * (2 * FHn)) return;
    int h = i / (2 * FHn), j = i % (2 * FHn);
    int k = (j < FHn) ? j : (j - FHn);
    float th = TWO_PI * (float)((h * k) & (Hn - 1)) / (float)Hn;
    float cv = cosf(th) * INV_SQRT_H, sv = sinf(th) * INV_SQRT_H;
    re[i] = (j < FHn) ? cv : -sv;
    im[i] = (j < FHn) ? sv :  cv;
}

// irfft along W, stacked-K table [258][256] (rows 0..128 hit Re, 129..257 hit Im):
//   x[w] = (1/16)[ Yre0 + (-1)^w Yre128 + 2 sum_{f=1..127}(Yre cos - Yim sin) ]
__global__ void build_wi(float* __restrict__ t)
{
    int i = blockIdx.x * 256 + threadIdx.x;
    if (i >= (2 * FWn) * Wn) return;
    int j = i / Wn, w = i % Wn;
    int f = (j <= 128) ? j : (j - FWn);
    float th = TWO_PI * (float)((w * f) & (Wn - 1)) / (float)Wn;
    float val;
    if (j <= 128) {
        float alpha = (f == 0 || f == 128) ? 1.f : 2.f;
        val = alpha * cosf(th) * INV_SQRT_W;
    } else {
        val = (f == 0 || f == 128) ? 0.f : (-2.f * sinf(th) * INV_SQRT_W);
    }
    t[i] = val;
}

// ---------------------------------------------------------------------------
// Host-side plumbing
// ---------------------------------------------------------------------------
static void launch_gemm(hipStream_t st,
                        const float* A, int lda, long sA,
                        const float* B, int ldb, long sB,
                        float* C, int ldc, long sC,
                        const float* bias, const float* res, long sR,
                        int M, int N, int K, int batch, bool gelu)
{
    dim3 grid((N + 127) / 128, (M + 15) / 16, batch);
    dim3 block(128);
    if (bias && res)
        gemm16<true, true, false><<<grid, block, 0, st>>>(A, lda, sA, B, ldb, sB, C, ldc, sC, bias, res, sR, M, N, K);
    else if (bias && gelu)
        gemm16<true, false, true><<<grid, block, 0, st>>>(A, lda, sA, B, ldb, sB, C, ldc, sC, bias, res, sR, M, N, K);
    else if (bias)
        gemm16<true, false, false><<<grid, block, 0, st>>>(A, lda, sA, B, ldb, sB, C, ldc, sC, bias, res, sR, M, N, K);
    else if (res)
        gemm16<false, true, false><<<grid, block, 0, st>>>(A, lda, sA, B, ldb, sB, C, ldc, sC, bias, res, sR, M, N, K);
    else
        gemm16<false, false, false><<<grid, block, 0, st>>>(A, lda, sA, B, ldb, sB, C, ldc, sC, bias, res, sR, M, N, K);
}

extern "C" void kernel_launch(void* const* d_in, const int* in_sizes, int n_in,
                              void* d_out, int out_size, void* d_ws, size_t ws_size,
                              hipStream_t stream)
{
    const float* x      = (const float*)d_in[0];   // [B,C,H,W]
    const float* proj_w = (const float*)d_in[1];   // [E,C]
    const float* proj_b = (const float*)d_in[2];   // [E]
    const float* fw_r   = (const float*)d_in[3];   // [NB,E,E,FH,FW]
    const float* fw_i   = (const float*)d_in[4];
    const float* mlp_w1 = (const float*)d_in[5];   // [4E,E]
    const float* mlp_b1 = (const float*)d_in[6];
    const float* mlp_w2 = (const float*)d_in[7];   // [E,4E]
    const float* mlp_b2 = (const float*)d_in[8];
    float* out = (float*)d_out;                    // [B,E,H,W]

    // carve workspace (floats) — total ~102 MB
    float* w = (float*)d_ws;
    size_t o = 0;
    auto grab = [&](size_t nf) { float* p = w + o; o += nf; return p; };
    const size_t PLANE = (size_t)Bn * En * HWn;          // 4,194,304
    float* buf0 = grab(PLANE);
    float* buf1 = grab(PLANE);
    float* XF1  = grab((size_t)Bn * En * 2 * Hn * FWn);  // [be][2H][FW]
    float* XF2  = grab((size_t)Bn * En * 2 * FREQS);     // [be][2][FH][FW]
    float* XM   = grab((size_t)Bn * En * 2 * FREQS);
    float* XF3  = grab((size_t)Bn * En * Hn * 2 * FWn);  // [be*H][258]
    float* hbuf = grab((size_t)4 * En * HWn);            // one batch of MLP hidden
    float* tWfc = grab((size_t)Wn * FWn);
    float* tWfs = grab((size_t)Wn * FWn);
    float* tHfr = grab((size_t)FHn * 2 * Hn);
    float* tHfi = grab((size_t)FHn * 2 * Hn);
    float* tHir = grab((size_t)Hn * 2 * FHn);
    float* tHii = grab((size_t)Hn * 2 * FHn);
    float* tWi  = grab((size_t)2 * FWn * Wn);

    // 0) build twiddle tables
    build_wf<<<(Wn * FWn + 255) / 256, 256, 0, stream>>>(tWfc, tWfs);
    build_hf<<<(FHn * 2 * Hn + 255) / 256, 256, 0, stream>>>(tHfr, tHfi);
    build_hi<<<(Hn * 2 * FHn + 255) / 256, 256, 0, stream>>>(tHir, tHii);
    build_wi<<<(2 * FWn * Wn + 255) / 256, 256, 0, stream>>>(tWi);

    // 1) input projection: per b: cur[e,hw] = proj_w[e,c] @ x[c,hw] + proj_b
    launch_gemm(stream, proj_w, Cn, 0,
                x, HWn, (long)Cn * HWn,
                buf0, HWn, (long)En * HWn,
                proj_b, nullptr, 0,
                En, HWn, Cn, Bn, false);

    float* cur = buf0;
    float* nxt = buf1;
    const long stXF1 = 2L * Hn * FWn;      // 33024
    const long stXF2 = 2L * FREQS;         // 16770
    const long stXF3 = (long)Hn * 2 * FWn; // 33024

    for (int i = 0; i < NBn; ++i) {
        // 2a) rfft along W (two real GEMMs), batch over be=128
        launch_gemm(stream, cur, Wn, (long)HWn,
                    tWfc, FWn, 0,
                    XF1, FWn, stXF1,
                    nullptr, nullptr, 0, Hn, FWn, Wn, Bn * En, false);
        launch_gemm(stream, cur, Wn, (long)HWn,
                    tWfs, FWn, 0,
                    XF1 + (long)Hn * FWn, FWn, stXF1,
                    nullptr, nullptr, 0, Hn, FWn, Wn, Bn * En, false);

        // 2b) fft along H, truncated to k<65 (stacked-K complex-as-real GEMMs)
        launch_gemm(stream, tHfr, 2 * Hn, 0,
                    XF1, FWn, stXF1,
                    XF2, FWn, stXF2,
                    nullptr, nullptr, 0, FHn, FWn, 2 * Hn, Bn * En, false);
        launch_gemm(stream, tHfi, 2 * Hn, 0,
                    XF1, FWn, stXF1,
                    XF2 + FREQS, FWn, stXF2,
                    nullptr, nullptr, 0, FHn, FWn, 2 * Hn, Bn * En, false);

        // 2c) per-frequency complex channel mix
        const size_t wofs = (size_t)i * En * En * FREQS;
        mix_kernel<<<FREQS, 128, 0, stream>>>(XF2, fw_r + wofs, fw_i + wofs, XM);

        // 2d) ifft along H (zero-padded spectrum), write row-major [be*H][Re|Im]
        launch_gemm(stream, tHir, 2 * FHn, 0,
                    XM, FWn, stXF2,
                    XF3, 2 * FWn, stXF3,
                    nullptr, nullptr, 0, Hn, FWn, 2 * FHn, Bn * En, false);
        launch_gemm(stream, tHii, 2 * FHn, 0,
                    XM, FWn, stXF2,
                    XF3 + FWn, 2 * FWn, stXF3,
                    nullptr, nullptr, 0, Hn, FWn, 2 * FHn, Bn * En, false);

        // 2e) irfft along W + residual: nxt[row][w] = cur + XF3[row] @ tWi
        launch_gemm(stream, XF3, 2 * FWn, 0,
                    tWi, Wn, 0,
                    nxt, Wn, 0,
                    nullptr, cur, 0,
                    Bn * En * Hn, Wn, 2 * FWn, 1, false);

        float* t = cur; cur = nxt; nxt = t;
    }

    // 3) MLP per batch image (reuse hbuf):
    for (int b = 0; b < Bn; ++b) {
        const float* xb = cur + (size_t)b * En * HWn;
        float* ob = out + (size_t)b * En * HWn;
        // h = gelu(W1 @ x + b1)   [128 x HW]
        launch_gemm(stream, mlp_w1, En, 0,
                    xb, HWn, 0,
                    hbuf, HWn, 0,
                    mlp_b1, nullptr, 0, 4 * En, HWn, En, 1, true);
        // out = x + W2 @ h + b2   [32 x HW]
        launch_gemm(stream, mlp_w2, 4 * En, 0,
                    hbuf, HWn, 0,
                    ob, HWn, 0,
                    mlp_b2, xb, 0, En, HWn, 4 * En, 1, false);
    }
}